// QConv1D_34565896798386
// MI455X (gfx1250) — compile-verified
//
#include <hip/hip_runtime.h>
#include <math.h>

typedef float v2f __attribute__((ext_vector_type(2)));
typedef float v8f __attribute__((ext_vector_type(8)));

#define NWIN   16        // windows per workgroup
#define DSTATE 512       // 2^9
#define STRIDE 516       // padded LDS row stride (floats) -> conflict-free A-frag loads
#define LOUT   2046
#define NCH    8
#define NQ     9

// ---- fp32 WMMA K-loop over 4 N-tiles (full 512 columns): s1 = E * s0 --------
// A frag (16x4 f32): lanes 0-15: M=lane, regs K=k+0,k+1 ; lanes 16-31: M=lane-16, K=k+2,k+3
// B frag (4x16 f32): lanes 0-15: N=lane, regs K=k+0,k+1 ; lanes 16-31: N=lane-16, K=k+2,k+3
// C/D  (16x16 f32): reg r, lanes 0-15: M=r, N=lane ; lanes 16-31: M=r+8, N=lane-16
__device__ __forceinline__ void wmma_k512_full(const float* __restrict__ srcLDS,
                                               const float* __restrict__ E,
                                               int lane, int wv, v8f acc[4]) {
  const int l16 = lane & 15;
  const int kh  = (lane >> 4) << 1;       // 0 or 2
  const int jt0 = wv << 2;                // 4 N-tiles per wave (32 tiles total)
  const float* a0 = srcLDS + l16 * STRIDE + kh;
  const float* e0 = E + (size_t)((jt0 + 0) * 16 + l16) * DSTATE + kh;
  const float* e1 = e0 + 16 * DSTATE;
  const float* e2 = e1 + 16 * DSTATE;
  const float* e3 = e2 + 16 * DSTATE;
#pragma unroll 4
  for (int k = 0; k < DSTATE; k += 4) {
    v2f a  = *(const v2f*)(a0 + k);
    v2f b0 = *(const v2f*)(e0 + k);
    v2f b1 = *(const v2f*)(e1 + k);
    v2f b2 = *(const v2f*)(e2 + k);
    v2f b3 = *(const v2f*)(e3 + k);
    acc[0] = __builtin_amdgcn_wmma_f32_16x16x4_f32(false, a, false, b0, (short)0, acc[0], false, false);
    acc[1] = __builtin_amdgcn_wmma_f32_16x16x4_f32(false, a, false, b1, (short)0, acc[1], false, false);
    acc[2] = __builtin_amdgcn_wmma_f32_16x16x4_f32(false, a, false, b2, (short)0, acc[2], false, false);
    acc[3] = __builtin_amdgcn_wmma_f32_16x16x4_f32(false, a, false, b3, (short)0, acc[3], false, false);
  }
}

// out_o(window) = s1^T (I_0 ⊗ M_1 ⊗ ... ⊗ M_8) s1,  M_i = [[sin t, cos t],[cos t, -sin t]]
// (exact identity: 2nd E-multiply + signed square-reduce == XOR-255 autocorrelation of r)
__global__ __launch_bounds__(256, 1)
void qconv1d_wmma_kernel(const float* __restrict__ x,     // (8,3,2048)
                         const float* __restrict__ E,     // (512,512)
                         const float* __restrict__ theta, // (8,9)
                         float* __restrict__ out) {       // (8,8,2046)
  __shared__ float bufA[NWIN * STRIDE];   // product state / butterfly scratch
  __shared__ float bufB[NWIN * STRIDE];   // s1 = E*s0 (preserved across channels)
  __shared__ float angC[NWIN][NQ];        // cos(x/2)
  __shared__ float angS[NWIN][NQ];        // sin(x/2)
  __shared__ float mC[NCH * NQ];          // cos(theta)  (full angle)
  __shared__ float mS[NCH * NQ];          // sin(theta)
  __shared__ float outAcc[NWIN];

  const int tid  = threadIdx.x;
  const int lane = tid & 31;
  const int wv   = tid >> 5;
  const int wg   = blockIdx.x;
  const int w16  = tid >> 4;              // window owned for butterfly phases
  const int t16  = tid & 15;

  // warm E into cache (emits global_prefetch)
  for (int r = tid; r < DSTATE; r += 256)
    __builtin_prefetch(E + (size_t)r * DSTATE, 0, 3);

  // ---- per-window half-angle tables for the product state ----
  if (tid < NWIN * NQ) {
    const int w = tid / NQ, i = tid % NQ;
    const int widx = wg * NWIN + w;
    const int b = widx / LOUT, l = widx % LOUT;
    const int ch = i / 3, kk = i % 3;
    const float a = x[(b * 3 + ch) * 2048 + l + kk];
    float sv, cv;
    sincosf(0.5f * a, &sv, &cv);
    angC[w][i] = cv;
    angS[w][i] = sv;
  }
  // ---- channel M-tables: FULL-angle cos/sin(theta) (theta[o][0] unused) ----
  if (tid < NCH * NQ) {
    float sv, cv;
    sincosf(theta[tid], &sv, &cv);
    mC[tid] = cv;
    mS[tid] = sv;
  }
  __syncthreads();

  // ---- product state s0 into bufA: 16 threads per window, 32 entries each ----
  {
    const int jbase = t16 << 5;
    float cc[NQ], ss[NQ];
#pragma unroll
    for (int i = 0; i < NQ; ++i) { cc[i] = angC[w16][i]; ss[i] = angS[w16][i]; }
    float base = 1.0f;
#pragma unroll
    for (int i = 0; i < 4; ++i)
      base *= ((t16 >> (3 - i)) & 1) ? ss[i] : cc[i];
#pragma unroll
    for (int jj = 0; jj < 32; ++jj) {
      float p = base;
#pragma unroll
      for (int i = 4; i < NQ; ++i)
        p *= ((jj >> (8 - i)) & 1) ? ss[i] : cc[i];
      bufA[w16 * STRIDE + jbase + jj] = p;
    }
  }
  __syncthreads();

  // ---- s1 = E * s0 : full fp32 WMMA GEMM, store into bufB ----
  {
    v8f acc[4] = {(v8f)0.f, (v8f)0.f, (v8f)0.f, (v8f)0.f};
    wmma_k512_full(bufA, E, lane, wv, acc);
    const int mb  = (lane >> 4) << 3;
    const int l16 = lane & 15;
    const int jt0 = wv << 2;
#pragma unroll
    for (int r = 0; r < 8; ++r) {
      bufB[(mb + r) * STRIDE + (jt0 + 0) * 16 + l16] = acc[0][r];
      bufB[(mb + r) * STRIDE + (jt0 + 1) * 16 + l16] = acc[1][r];
      bufB[(mb + r) * STRIDE + (jt0 + 2) * 16 + l16] = acc[2][r];
      bufB[(mb + r) * STRIDE + (jt0 + 3) * 16 + l16] = acc[3][r];
    }
  }
  __syncthreads();

  // ---- per channel: t = (⊗M_i) s1 via register butterflies, out = <s1, t> ----
  for (int o = 0; o < NCH; ++o) {
    const float* tc = &mC[o * NQ];
    const float* ts = &mS[o * NQ];
    if (tid < NWIN) outAcc[tid] = 0.0f;

    // Phase 1: thread owns j = t16 + 16k (k=0..31); j bit 4+b == k bit b.
    // Qubit i (1..4) acts on j bit 8-i == k bit 4-i. Reads bufB, writes bufA.
    {
      const float* S = bufB + w16 * STRIDE + t16;
      float* Dp = bufA + w16 * STRIDE + t16;
      float v[32];
#pragma unroll
      for (int k = 0; k < 32; ++k) v[k] = S[16 * k];
#pragma unroll
      for (int i = 1; i <= 4; ++i) {
        const int bb = 1 << (4 - i);
        const float c = tc[i], s = ts[i];
#pragma unroll
        for (int k0 = 0; k0 < 32; ++k0) {
          if (k0 & bb) continue;
          const int k1 = k0 | bb;
          const float a0 = v[k0], a1 = v[k1];
          v[k0] = s * a0 + c * a1;     // M = [[sin, cos],[cos, -sin]]
          v[k1] = c * a0 - s * a1;
        }
      }
#pragma unroll
      for (int k = 0; k < 32; ++k) Dp[16 * k] = v[k];
    }
    __syncthreads();

    // Phase 2: thread owns j = 32*t16 + jj; qubit i (5..8) acts on jj bit 8-i.
    // Finish t in registers, fuse the dot product with s1 (bufB), no store.
    {
      const float* S = bufA + w16 * STRIDE + (t16 << 5);
      const float* S1 = bufB + w16 * STRIDE + (t16 << 5);
      float v[32];
#pragma unroll
      for (int jj = 0; jj < 32; ++jj) v[jj] = S[jj];
#pragma unroll
      for (int i = 5; i <= 8; ++i) {
        const int bb = 1 << (8 - i);
        const float c = tc[i], s = ts[i];
#pragma unroll
        for (int j0 = 0; j0 < 32; ++j0) {
          if (j0 & bb) continue;
          const int j1 = j0 | bb;
          const float a0 = v[j0], a1 = v[j1];
          v[j0] = s * a0 + c * a1;
          v[j1] = c * a0 - s * a1;
        }
      }
      float dot = 0.0f;
#pragma unroll
      for (int jj = 0; jj < 32; ++jj) dot += v[jj] * S1[jj];
      atomicAdd(&outAcc[w16], dot);    // ds_add_f32
    }
    __syncthreads();

    if (tid < NWIN) {
      const int widx = wg * NWIN + tid;
      const int b = widx / LOUT, l = widx % LOUT;
      out[(b * NCH + o) * LOUT + l] = outAcc[tid];
    }
    __syncthreads();
  }
}

extern "C" void kernel_launch(void* const* d_in, const int* in_sizes, int n_in,
                              void* d_out, int out_size, void* d_ws, size_t ws_size,
                              hipStream_t stream) {
  (void)in_sizes; (void)n_in; (void)out_size; (void)d_ws; (void)ws_size;
  const float* x     = (const float*)d_in[0];
  const float* E     = (const float*)d_in[1];
  const float* theta = (const float*)d_in[2];
  float* out         = (float*)d_out;
  // N = 8*2046 = 16368 windows = 1023 tiles of 16
  hipLaunchKernelGGL(qconv1d_wmma_kernel, dim3(1023), dim3(256), 0, stream,
                     x, E, theta, out);
}